// MultiheadLinearAttention_33423435498097
// MI455X (gfx1250) — compile-verified
//
#include <hip/hip_runtime.h>
#include <hip/hip_bf16.h>
#include <cstdint>

typedef __attribute__((ext_vector_type(2))) float v2f;
typedef __attribute__((ext_vector_type(8))) float v8f;

#define S_ 2048
#define B_ 4
#define E_ 1024
#define H_ 16
#define P_ 256
#define T_ 2048
#define KCH 8
#define KCHUNK (S_ / KCH) /* 256 */

// -------- Stage 0: sbar[h] = (1/P) * sum_s cl_f[h*2048 + s] --------
__global__ __launch_bounds__(512) void sbar_kernel(const float* __restrict__ cl_f,
                                                   float* __restrict__ sbar) {
  const int h = threadIdx.x >> 5;
  const int lane = threadIdx.x & 31;
  float acc = 0.f;
  for (int j = lane; j < S_; j += 32) acc += cl_f[h * S_ + j];
  for (int off = 16; off > 0; off >>= 1) acc += __shfl_xor(acc, off, 32);
  if (lane == 0) sbar[h] = acc * (1.0f / P_);
}

// -------- Stage 1: agg partials via V_WMMA_F32_16X16X4_F32 --------
// part[kc][b][h][e] = sum_{s in chunk kc} w_f[h,s] * value[s,b,e]
// w_f[h,s] == cl_f_flat[h*2048 + s]   (since P*C == S and s = p*C + c)
__global__ __launch_bounds__(256) void agg_partial_kernel(
    const float* __restrict__ value, const float* __restrict__ cl_f,
    float* __restrict__ part) {
  __shared__ float wt[KCHUNK * H_];  // 16 KB, transposed: wt[s*16 + h]
  const int tid = threadIdx.x;
  const int kc = blockIdx.x & 7;
  const int ng = (blockIdx.x >> 3) & 7;
  const int b  = blockIdx.x >> 6;
  const int s0 = kc * KCHUNK;

  // Cooperative, coalesced stage of the cl_f K-chunk (transposed for
  // conflict-free A-fragment reads below).
  for (int i = tid; i < KCHUNK * H_; i += 256) {
    const int h = i >> 8;            // i / 256
    const int s = i & (KCHUNK - 1);  // i % 256
    wt[s * H_ + h] = cl_f[h * S_ + s0 + s];
  }
  __syncthreads();

  const int lane = tid & 31;
  const int wave = tid >> 5;
  const int n0 = (ng * 8 + wave) * 16;  // E-column tile base
  const int m  = lane & 15;             // A: row M(=h); B: col N offset
  const int kb = (lane >> 4) * 2;       // lane-half selects K {0,1} or {2,3}

  const float* vb = value + (size_t)s0 * (B_ * E_) + (size_t)b * E_ + n0 + m;

  v8f acc = {};
  for (int s = 0; s < KCHUNK; s += 4) {
    v2f a, bm;
    // A fragment: 16x4 f32, A[m, k] = w_f[m, s0+s+k]
    a.x = wt[(s + kb) * H_ + m];
    a.y = wt[(s + kb + 1) * H_ + m];
    // B fragment: 4x16 f32, B[k, n] = value[s0+s+k, b, n0+n]
    bm.x = vb[(size_t)(s + kb) * (B_ * E_)];
    bm.y = vb[(size_t)(s + kb + 1) * (B_ * E_)];
    acc = __builtin_amdgcn_wmma_f32_16x16x4_f32(
        /*neg_a=*/false, a, /*neg_b=*/false, bm,
        /*c_mod=*/(short)0, acc, /*reuse_a=*/false, /*reuse_b=*/false);
  }

  // C/D layout: VGPR v, lane l -> (M = v + 8*(l>>4), N = l&15)
  const int hb = (lane >> 4) * 8;
  const int e  = n0 + (lane & 15);
  float* dst = part + (size_t)kc * (B_ * H_ * E_) + (size_t)b * (H_ * E_) + e;
#pragma unroll
  for (int v = 0; v < 8; ++v) dst[(hb + v) * E_] = acc[v];
}

// -------- Stage 2: reduce K-split partials, apply 1/P --------
__global__ __launch_bounds__(256) void agg_reduce_kernel(
    const float* __restrict__ part, float* __restrict__ agg) {
  const int i = blockIdx.x * 256 + threadIdx.x;
  float a = 0.f;
#pragma unroll
  for (int kc = 0; kc < KCH; ++kc) a += part[(size_t)kc * (B_ * H_ * E_) + i];
  agg[i] = a * (1.0f / P_);
}

// -------- Stage 3: vbar[b,e] = v_w[e,:].agg[b,e>>6,:] + v_b[e]*sbar[e>>6] ---
__global__ __launch_bounds__(256) void vbar_kernel(
    const float* __restrict__ v_w, const float* __restrict__ v_b,
    const float* __restrict__ agg, const float* __restrict__ sbar,
    float* __restrict__ vbar) {
  const int lane = threadIdx.x & 31;
  const int e = blockIdx.x * 8 + (threadIdx.x >> 5);
  const int h = e >> 6;
  const float* w  = v_w + (size_t)e * E_;
  const float* a0 = agg + (size_t)(0 * H_ + h) * E_;
  const float* a1 = agg + (size_t)(1 * H_ + h) * E_;
  const float* a2 = agg + (size_t)(2 * H_ + h) * E_;
  const float* a3 = agg + (size_t)(3 * H_ + h) * E_;
  float c0 = 0.f, c1 = 0.f, c2 = 0.f, c3 = 0.f;
  for (int j = lane; j < E_; j += 32) {
    const float ww = w[j];
    c0 += ww * a0[j]; c1 += ww * a1[j]; c2 += ww * a2[j]; c3 += ww * a3[j];
  }
  for (int off = 16; off > 0; off >>= 1) {
    c0 += __shfl_xor(c0, off, 32); c1 += __shfl_xor(c1, off, 32);
    c2 += __shfl_xor(c2, off, 32); c3 += __shfl_xor(c3, off, 32);
  }
  if (lane == 0) {
    const float bias = v_b[e] * sbar[h];
    vbar[0 * E_ + e] = c0 + bias; vbar[1 * E_ + e] = c1 + bias;
    vbar[2 * E_ + e] = c2 + bias; vbar[3 * E_ + e] = c3 + bias;
  }
}

// -------- Stage 4: ob[b,eo] = o_w[eo,:].vbar[b,:] + o_b[eo] --------
__global__ __launch_bounds__(256) void ob_kernel(
    const float* __restrict__ o_w, const float* __restrict__ o_b,
    const float* __restrict__ vbar, float* __restrict__ ob) {
  const int lane = threadIdx.x & 31;
  const int eo = blockIdx.x * 8 + (threadIdx.x >> 5);
  const float* w = o_w + (size_t)eo * E_;
  float c0 = 0.f, c1 = 0.f, c2 = 0.f, c3 = 0.f;
  for (int j = lane; j < E_; j += 32) {
    const float ww = w[j];
    c0 += ww * vbar[0 * E_ + j]; c1 += ww * vbar[1 * E_ + j];
    c2 += ww * vbar[2 * E_ + j]; c3 += ww * vbar[3 * E_ + j];
  }
  for (int off = 16; off > 0; off >>= 1) {
    c0 += __shfl_xor(c0, off, 32); c1 += __shfl_xor(c1, off, 32);
    c2 += __shfl_xor(c2, off, 32); c3 += __shfl_xor(c3, off, 32);
  }
  if (lane == 0) {
    const float bb = o_b[eo];
    ob[0 * E_ + eo] = c0 + bb; ob[1 * E_ + eo] = c1 + bb;
    ob[2 * E_ + eo] = c2 + bb; ob[3 * E_ + eo] = c3 + bb;
  }
}

// -------- Stage 5: broadcast ob (B*E) over all T rows, float4 streams ------
__global__ __launch_bounds__(256) void bcast_kernel(
    const float4* __restrict__ ob4, float4* __restrict__ out4) {
  __shared__ float4 l[(B_ * E_) / 4];  // 1024 float4 = 16 KB
  for (int i = threadIdx.x; i < (B_ * E_) / 4; i += 256) l[i] = ob4[i];
  __syncthreads();
  const size_t base = (size_t)blockIdx.x * ((B_ * E_) / 4);
#pragma unroll
  for (int k = 0; k < 4; ++k) {
    const int j = threadIdx.x + k * 256;
    out4[base + j] = l[j];
  }
}

extern "C" void kernel_launch(void* const* d_in, const int* in_sizes, int n_in,
                              void* d_out, int out_size, void* d_ws, size_t ws_size,
                              hipStream_t stream) {
  // Input order: query,key,value,q_w,q_b,k_w,k_b,v_w,v_b,o_w,o_b,cl_e,cl_f,conv_w0..4
  // Q/K/score/conv inputs are mathematically dead (softmax of a p-constant).
  const float* value = (const float*)d_in[2];
  const float* v_w   = (const float*)d_in[7];
  const float* v_b   = (const float*)d_in[8];
  const float* o_w   = (const float*)d_in[9];
  const float* o_b   = (const float*)d_in[10];
  const float* cl_f  = (const float*)d_in[12];

  float* ws   = (float*)d_ws;
  float* part = ws;                                   // 8*4*16*1024 = 524288 f
  float* agg  = part + (size_t)KCH * B_ * H_ * E_;    // 65536 f
  float* sbar = agg + (size_t)B_ * H_ * E_;           // 16 f
  float* vbar = sbar + 16;                            // 4096 f
  float* obuf = vbar + (size_t)B_ * E_;               // 4096 f (16B aligned)

  sbar_kernel<<<1, 512, 0, stream>>>(cl_f, sbar);
  agg_partial_kernel<<<B_ * 8 * KCH, 256, 0, stream>>>(value, cl_f, part);
  agg_reduce_kernel<<<(B_ * H_ * E_) / 256, 256, 0, stream>>>(part, agg);
  vbar_kernel<<<E_ / 8, 256, 0, stream>>>(v_w, v_b, agg, sbar, vbar);
  ob_kernel<<<E_ / 8, 256, 0, stream>>>(o_w, o_b, vbar, obuf);
  bcast_kernel<<<(T_ * B_ * E_) / (B_ * E_), 256, 0, stream>>>(
      (const float4*)obuf, (float4*)d_out);

  (void)in_sizes; (void)n_in; (void)out_size; (void)ws_size;
}